// DynMoleRouterLoss_76063870812350
// MI455X (gfx1250) — compile-verified
//
#include <hip/hip_runtime.h>
#include <hip/hip_bf16.h>

// DynMole router loss on gfx1250 (MI455X), compile-only tuned.
// Stage 1: grid-stride over tokens, per-lane softmax + rank/top-p mask,
//          per-block deterministic reduction; cross-lane per-expert sums via
//          V_WMMA_F32_16X16X4_F32 (A = grouped partials, B = ones).
// Stage 2: single-wave deterministic reduction of per-block partials.

typedef __attribute__((ext_vector_type(2))) float v2f;
typedef __attribute__((ext_vector_type(4))) float v4f;
typedef __attribute__((ext_vector_type(8))) float v8f;

#define DM_TOPP 0.8f
#define DM_KEEPK 2
#define DM_ENTTH 0.5f
#define DM_NCOMP 18           // 0..7 routing, 8..15 probs, 16 ent, 17 tokmask
#define DM_BLOCK 256
#define DM_NBMAX 2048

__global__ __launch_bounds__(DM_BLOCK) void dynmole_stage1(
    const float* __restrict__ gate, const int* __restrict__ amask,
    float* __restrict__ part, int T, int M, int NB)
{
    __shared__ float lpart[DM_BLOCK * DM_NCOMP];
    __shared__ float l2[32 * DM_NCOMP];

    const int tid = threadIdx.x;
    const int gsz = NB * DM_BLOCK;

    float accR[8], accP[8];
#pragma unroll
    for (int e = 0; e < 8; ++e) { accR[e] = 0.0f; accP[e] = 0.0f; }
    float accE = 0.0f, accM = 0.0f;

    for (int t = blockIdx.x * DM_BLOCK + tid; t < T; t += gsz) {
        const v4f* g4 = (const v4f*)gate + (size_t)t * 2;
        // next-iteration prefetch (global_prefetch_b8 path)
        if (t + gsz < T)
            __builtin_prefetch((const v4f*)gate + (size_t)(t + gsz) * 2, 0, 0);

        v4f x0 = __builtin_nontemporal_load(&g4[0]);
        v4f x1 = __builtin_nontemporal_load(&g4[1]);
        float x[8] = {x0.x, x0.y, x0.z, x0.w, x1.x, x1.y, x1.z, x1.w};

        float tm = (float)amask[t % M];

        // softmax over 8
        float mx = x[0];
#pragma unroll
        for (int i = 1; i < 8; ++i) mx = fmaxf(mx, x[i]);
        float e8[8], s = 0.0f;
#pragma unroll
        for (int i = 0; i < 8; ++i) { e8[i] = __expf(x[i] - mx); s += e8[i]; }
        float inv = 1.0f / s;
        float p[8], sq = 0.0f;
#pragma unroll
        for (int i = 0; i < 8; ++i) { p[i] = e8[i] * inv; sq += p[i] * p[i]; }

        float ent = 1.0f - sq;               // Tsallis q=2
        bool hot = ent > DM_ENTTH;

        // rank + inclusive sorted-cumsum per element (stable tie-break j<i)
#pragma unroll
        for (int i = 0; i < 8; ++i) {
            float c = p[i];
            int r = 0;
#pragma unroll
            for (int j = 0; j < 8; ++j) {
                if (j == i) continue;
                bool before = (j < i) ? (p[j] >= p[i]) : (p[j] > p[i]);
                r += before ? 1 : 0;
                c += before ? p[j] : 0.0f;
            }
            float mi = (hot || (r < DM_KEEPK) || (c <= DM_TOPP)) ? 1.0f : 0.0f;
            accR[i] += p[i] * mi * tm;
            accP[i] += p[i] * tm;
        }
        accE += ent;
        accM += tm;
    }

    // dump per-thread accumulators to LDS
#pragma unroll
    for (int e = 0; e < 8; ++e) {
        lpart[tid * DM_NCOMP + e]     = accR[e];
        lpart[tid * DM_NCOMP + 8 + e] = accP[e];
    }
    lpart[tid * DM_NCOMP + 16] = accE;
    lpart[tid * DM_NCOMP + 17] = accM;
    __syncthreads();

    if (tid < 32) {                          // wave 0 only: EXEC all-1s here
        const int L = tid;
        // fixed-order cross-wave reduction (deterministic)
        float s18[DM_NCOMP];
#pragma unroll
        for (int c = 0; c < DM_NCOMP; ++c) {
            float v = 0.0f;
#pragma unroll
            for (int w = 0; w < DM_BLOCK / 32; ++w)
                v += lpart[(L + 32 * w) * DM_NCOMP + c];
            s18[c] = v;
        }
#pragma unroll
        for (int c = 0; c < DM_NCOMP; ++c) l2[L * DM_NCOMP + c] = s18[c];
        // same-wave LDS ops are in-order; compiler inserts dscnt waits on use.

        // Build A (16x4 f32): row m = expert component m, K = 4 lane-group sums.
        // Layout: lanes0-15 hold K0(v0)/K1(v1); lanes16-31 hold K2(v0)/K3(v1).
        const int comp = L & 15;
        const int base = (L < 16) ? 0 : 16;
        float a0 = 0.0f, a1 = 0.0f;
#pragma unroll
        for (int l = 0; l < 8; ++l) {
            a0 += l2[(base + l) * DM_NCOMP + comp];
            a1 += l2[(base + 8 + l) * DM_NCOMP + comp];
        }
        v2f A; A.x = a0; A.y = a1;
        v2f B; B.x = 1.0f; B.y = 1.0f;       // ones: D[m][n] = sum_k A[m][k]
        v8f C = {};
        v8f D = __builtin_amdgcn_wmma_f32_16x16x4_f32(
            false, A, false, B, (short)0, C, false, false);

        // ent / tokmask cross-lane reduce
        float entT = s18[16], mT = s18[17];
#pragma unroll
        for (int off = 16; off > 0; off >>= 1) {
            entT += __shfl_xor(entT, off, 32);
            mT   += __shfl_xor(mT, off, 32);
        }

        float* bp = part + (size_t)blockIdx.x * DM_NCOMP;
        if (L == 0) {                        // N=0 column, M=0..7 in D[0..7]
#pragma unroll
            for (int r = 0; r < 8; ++r) bp[r] = D[r];
            bp[16] = entT;
            bp[17] = mT;
        } else if (L == 16) {                // N=0 column, M=8..15 in D[0..7]
#pragma unroll
            for (int r = 0; r < 8; ++r) bp[8 + r] = D[r];
        }
    }
}

__global__ __launch_bounds__(32) void dynmole_stage2(
    const float* __restrict__ part, float* __restrict__ out, int NB, float invT)
{
    const int L = threadIdx.x;
    float tot = 0.0f;
    if (L < DM_NCOMP)
        for (int b = 0; b < NB; ++b)         // fixed order: deterministic
            tot += part[(size_t)b * DM_NCOMP + L];

    float denom = __shfl(tot, 17, 32) + 1e-8f;
    float entMean = __shfl(tot, 16, 32) * invT;
    float lb = 0.0f;
#pragma unroll
    for (int e = 0; e < 8; ++e) {
        float R = __shfl(tot, e, 32);
        float P = __shfl(tot, 8 + e, 32);
        lb += R * P;
    }
    lb = 8.0f * lb / (denom * denom);
    if (L == 0) out[0] = 0.01f * entMean + 0.001f * lb;
}

extern "C" void kernel_launch(void* const* d_in, const int* in_sizes, int n_in,
                              void* d_out, int out_size, void* d_ws, size_t ws_size,
                              hipStream_t stream) {
    const float* gate = (const float*)d_in[0];   // [T, 8] f32
    const int*   am   = (const int*)d_in[1];     // [B*S] i32
    const int T = in_sizes[0] / 8;
    const int M = in_sizes[1];

    int NB = DM_NBMAX;
    size_t need = (size_t)NB * DM_NCOMP * sizeof(float);
    if (ws_size < need) {
        NB = (int)(ws_size / (DM_NCOMP * sizeof(float)));
        if (NB < 1) NB = 1;
    }
    float* part = (float*)d_ws;

    dynmole_stage1<<<NB, DM_BLOCK, 0, stream>>>(gate, am, part, T, M, NB);
    dynmole_stage2<<<1, 32, 0, stream>>>(part, (float*)d_out, NB, 1.0f / (float)T);
}